// MultiNeRFBlock_25245817765917
// MI455X (gfx1250) — compile-verified
//
#include <hip/hip_runtime.h>
#include <hip/hip_bf16.h>

typedef float v2f __attribute__((ext_vector_type(2)));
typedef float v8f __attribute__((ext_vector_type(8)));

#define IN_DIM   256
#define OUT_DIM  256
#define RANK     32
#define NTILE    16              // samples per tile (WMMA N)
#define KSTEPS   (IN_DIM / 4)    // 64 wmma_f32_16x16x4 steps
#define LDS_STRIDE 260           // 260 % 64 == 4 -> conflict-free b64 reads
#define WAVES    16
#define BLOCK    (WAVES * 32)
#define NEG_SLOPE 0.01f

// ---------------------------------------------------------------------------
// Kernel 1: fold the low-rank adapter into the dense weights.
//   Wmod[o][i] = W[o][i] + sum_r A[task][i][r] * B[task][r][o]
// 256 blocks (one per output o), 256 threads (one per input i).
// ---------------------------------------------------------------------------
__global__ void build_weff_kernel(const float* __restrict__ W,
                                  const float* __restrict__ A,
                                  const float* __restrict__ B,
                                  const int* __restrict__ task_p,
                                  float* __restrict__ Wmod) {
  const int task = *task_p;
  const float* At = A + (size_t)task * IN_DIM * RANK;   // [256][32]
  const float* Bt = B + (size_t)task * RANK * OUT_DIM;  // [32][256]

  const int o = blockIdx.x;
  const int i = threadIdx.x;

  __shared__ float bcol[RANK];
  if (i < RANK) bcol[i] = Bt[i * OUT_DIM + o];
  __syncthreads();

  float s = W[o * IN_DIM + i];
  const float* arow = At + i * RANK;
#pragma unroll
  for (int r = 0; r < RANK; ++r) s += arow[r] * bcol[r];
  Wmod[o * IN_DIM + i] = s;
}

// ---------------------------------------------------------------------------
// Kernel 2: y = leaky_relu(x @ Wmod^T + b) via V_WMMA_F32_16X16X4_F32.
// 16 waves per WG; wave w owns output tile o in [16w, 16w+16).
// W_eff strip held in registers (64 A-fragments = 128 VGPRs, loaded once).
// x tiles staged global->LDS, double-buffered, one barrier per tile.
// ---------------------------------------------------------------------------
__global__ __launch_bounds__(BLOCK, 1) void fused_lora_gemm_kernel(
    const float* __restrict__ x, const float* __restrict__ Wmod,
    const float* __restrict__ bias, float* __restrict__ y, int ntiles) {
  __shared__ float ldsx[2][NTILE * LDS_STRIDE];

  const int tid  = threadIdx.x;
  const int lane = tid & 31;
  const int wave = tid >> 5;
  const int lm   = lane & 15;   // A: row M(o);  B/D: col N(n)
  const int h    = lane >> 4;   // lane half

  // --- A fragments: W_eff[o][k], lane -> o = 16*wave + lm, k = 4*kt + 2h + v
  const int oa = wave * NTILE + lm;
  const float* wrow = Wmod + oa * IN_DIM + 2 * h;
  v2f wf[KSTEPS];
#pragma unroll
  for (int kt = 0; kt < KSTEPS; ++kt)
    wf[kt] = *(const v2f*)(wrow + 4 * kt);

  // --- bias for D lanes: o = 16*wave + 8*h + v, v = 0..7
  const int ob = wave * NTILE + 8 * h;
  float bv[8];
#pragma unroll
  for (int v = 0; v < 8; ++v) bv[v] = bias[ob + v];

  // --- x-tile staging map: 1024 float4 per tile, 2 per thread (coalesced)
  const int idx0 = tid;
  const int idx1 = tid + BLOCK;
  const int r0 = idx0 >> 6, c0 = (idx0 & 63) << 2;
  const int r1 = idx1 >> 6, c1 = (idx1 & 63) << 2;

  int tile = blockIdx.x;
  if (tile >= ntiles) return;

  float4 p0 = *(const float4*)(x + (size_t)(tile * NTILE + r0) * IN_DIM + c0);
  float4 p1 = *(const float4*)(x + (size_t)(tile * NTILE + r1) * IN_DIM + c1);

  int buf = 0;
  for (; tile < ntiles; tile += gridDim.x) {
    // stage current tile into LDS (scalar stores; rows keep 8B alignment w/ pad 260)
    float* s0 = &ldsx[buf][r0 * LDS_STRIDE + c0];
    s0[0] = p0.x; s0[1] = p0.y; s0[2] = p0.z; s0[3] = p0.w;
    float* s1 = &ldsx[buf][r1 * LDS_STRIDE + c1];
    s1[0] = p1.x; s1[1] = p1.y; s1[2] = p1.z; s1[3] = p1.w;
    __syncthreads();

    // prefetch next tile while computing (lands in other buffer next iter)
    const int next = tile + gridDim.x;
    if (next < ntiles) {
      p0 = *(const float4*)(x + (size_t)(next * NTILE + r0) * IN_DIM + c0);
      p1 = *(const float4*)(x + (size_t)(next * NTILE + r1) * IN_DIM + c1);
    }

    // --- K loop: B frag = x_lds[n][4kt + 2h + v]; dual accumulators for ILP
    const float* bbase = &ldsx[buf][lm * LDS_STRIDE + 2 * h];
    v8f acc0 = {};
    v8f acc1 = {};
#pragma unroll
    for (int kt = 0; kt < KSTEPS; kt += 2) {
      v2f bf0 = *(const v2f*)(bbase + 4 * kt);
      acc0 = __builtin_amdgcn_wmma_f32_16x16x4_f32(
          false, wf[kt], false, bf0, (short)0, acc0, false, false);
      v2f bf1 = *(const v2f*)(bbase + 4 * (kt + 1));
      acc1 = __builtin_amdgcn_wmma_f32_16x16x4_f32(
          false, wf[kt + 1], false, bf1, (short)0, acc1, false, false);
    }

    // --- epilogue: bias + LeakyReLU, contiguous 32B per lane
    const int n = tile * NTILE + lm;
    float* yp = y + (size_t)n * OUT_DIM + ob;
    float out[8];
#pragma unroll
    for (int v = 0; v < 8; ++v) {
      float t = acc0[v] + acc1[v] + bv[v];
      out[v] = (t >= 0.f) ? t : NEG_SLOPE * t;
    }
    *(float4*)(yp)     = make_float4(out[0], out[1], out[2], out[3]);
    *(float4*)(yp + 4) = make_float4(out[4], out[5], out[6], out[7]);

    buf ^= 1;  // safe with the single barrier: writes to buf^1 while buf is read
  }
}

// ---------------------------------------------------------------------------
// Host launcher. Inputs (setup_inputs order): x, task, W, b, A, B.
// ---------------------------------------------------------------------------
extern "C" void kernel_launch(void* const* d_in, const int* in_sizes, int n_in,
                              void* d_out, int out_size, void* d_ws, size_t ws_size,
                              hipStream_t stream) {
  const float* x    = (const float*)d_in[0];
  const int*   task = (const int*)d_in[1];
  const float* W    = (const float*)d_in[2];
  const float* b    = (const float*)d_in[3];
  const float* A    = (const float*)d_in[4];
  const float* B    = (const float*)d_in[5];
  float*       y    = (float*)d_out;
  float*       Wmod = (float*)d_ws;  // 256*256*4 = 256 KB scratch

  const int N = in_sizes[0] / IN_DIM;
  const int ntiles = N / NTILE;

  build_weff_kernel<<<OUT_DIM, IN_DIM, 0, stream>>>(W, A, B, task, Wmod);

  int grid = ntiles < 1024 ? ntiles : 1024;
  fused_lora_gemm_kernel<<<grid, BLOCK, 0, stream>>>(x, Wmod, b, y, ntiles);
}